// LightGCN_37048387895609
// MI455X (gfx1250) — compile-verified
//
#include <hip/hip_runtime.h>
#include <hip/hip_bf16.h>

// LightGCN on gfx1250:
//  - propagation: wave-per-edge scatter with global_atomic_add_f32 (L2-resident
//    working set: ~90MB < 192MB L2), coalesced 256B row gathers, row prefetch.
//  - final batched dot: V_WMMA_F32_16X16X4_F32 (full f32 precision), diag extract.

typedef __attribute__((ext_vector_type(2))) float v2f;
typedef __attribute__((ext_vector_type(8))) float v8f;

#define DIM 64

// cur = acc = concat(user_w, item_w); next = 0; deg = 0
__global__ __launch_bounds__(256) void k_init(
    const float4* __restrict__ uw4, const float4* __restrict__ iw4,
    float4* __restrict__ cur4, float4* __restrict__ acc4, float4* __restrict__ nxt4,
    float4* __restrict__ deg4, int nu4, int total4, int ndeg4) {
  int i = blockIdx.x * blockDim.x + threadIdx.x;
  float4 z; z.x = 0.f; z.y = 0.f; z.z = 0.f; z.w = 0.f;
  if (i < total4) {
    float4 s = (i < nu4) ? uw4[i] : iw4[i - nu4];
    cur4[i] = s;
    acc4[i] = s;
    nxt4[i] = z;
  }
  if (i < ndeg4) deg4[i] = z;
}

// deg[row[e]] += 1   (exact: integer-valued f32 counts, order-independent)
__global__ __launch_bounds__(256) void k_deg(const int* __restrict__ row,
                                             float* __restrict__ deg, int E) {
  int i = blockIdx.x * blockDim.x + threadIdx.x;
  if (i < E) atomicAdd(&deg[row[i]], 1.0f);
}

// deg -> deg^(-1/2) in place (0 where deg == 0)
__global__ __launch_bounds__(256) void k_dis(float* __restrict__ deg, int N) {
  int i = blockIdx.x * blockDim.x + threadIdx.x;
  if (i < N) {
    float d = deg[i];
    deg[i] = (d > 0.f) ? (1.0f / sqrtf(d)) : 0.0f;
  }
}

// norm[e] = dis[row[e]] * dis[col[e]]
__global__ __launch_bounds__(256) void k_norm(const int* __restrict__ row,
                                              const int* __restrict__ col,
                                              const float* __restrict__ dis,
                                              float* __restrict__ norm, int E) {
  int i = blockIdx.x * blockDim.x + threadIdx.x;
  if (i < E) norm[i] = dis[row[i]] * dis[col[i]];
}

// nxt[col[e]] += norm[e] * cur[row[e]]  — one wave per edge iteration,
// 2 floats per lane (coalesced 256B row), f32 atomics into L2.
__global__ __launch_bounds__(256) void k_prop(const int* __restrict__ row,
                                              const int* __restrict__ col,
                                              const float* __restrict__ norm,
                                              const float* __restrict__ cur,
                                              float* __restrict__ nxt, int E) {
  int lane  = threadIdx.x & 31;
  int wave  = (int)((blockIdx.x * blockDim.x + threadIdx.x) >> 5);
  int nwav  = (int)((gridDim.x * blockDim.x) >> 5);
  for (int e = wave; e < E; e += nwav) {
    int eu = __builtin_amdgcn_readfirstlane(e);   // force scalar edge loads
    int r  = row[eu];
    int c  = col[eu];
    float w = norm[eu];
    int en = eu + nwav;                            // prefetch next source row
    if (en < E) {
      int rn = row[en];
      __builtin_prefetch((const void*)(cur + (size_t)rn * DIM), 0, 3);
    }
    const v2f* src = (const v2f*)(cur + (size_t)r * DIM);
    v2f v = src[lane];
    float* d = nxt + (size_t)c * DIM + lane * 2;
    atomicAdd(d,     w * v.x);
    atomicAdd(d + 1, w * v.y);
  }
}

// acc += nxt; zero old cur (becomes next layer's scatter target)
__global__ __launch_bounds__(256) void k_fuse(float4* __restrict__ acc4,
                                              const float4* __restrict__ nxt4,
                                              float4* __restrict__ cur4, int total4) {
  int i = blockIdx.x * blockDim.x + threadIdx.x;
  if (i < total4) {
    float4 a = acc4[i];
    float4 n = nxt4[i];
    a.x += n.x; a.y += n.y; a.z += n.z; a.w += n.w;
    acc4[i] = a;
    float4 z; z.x = 0.f; z.y = 0.f; z.z = 0.f; z.w = 0.f;
    cur4[i] = z;
  }
}

// out[b] = (acc_u . acc_i) / 16 via V_WMMA_F32_16X16X4_F32, diagonal extract.
// One wave computes a 16x16 tile over K=64 in 16 k-steps (full f32 precision).
__global__ __launch_bounds__(256) void k_dot(const float* __restrict__ acc,
                                             const int* __restrict__ users,
                                             const int* __restrict__ items,
                                             float* __restrict__ out,
                                             int NU, int Bn) {
  int wave = (int)((blockIdx.x * blockDim.x + threadIdx.x) >> 5);
  int lane = threadIdx.x & 31;
  int tiles = (Bn + 15) >> 4;
  if (wave >= tiles) return;               // wave-uniform: EXEC stays all-1s

  int n  = lane & 15;                      // A: M=n ; B: N=n (same addressing)
  int hi = lane >> 4;                      // lanes 16-31 carry K = {2,3} of each step
  int bb = wave * 16 + n;
  if (bb > Bn - 1) bb = Bn - 1;            // branchless clamp keeps EXEC full
  const float* urow = acc + (size_t)users[bb] * DIM;
  const float* irow = acc + ((size_t)NU + (size_t)items[bb]) * DIM;

  v8f c = {};
#pragma unroll
  for (int k0 = 0; k0 < DIM; k0 += 4) {
    v2f a = *(const v2f*)(urow + k0 + 2 * hi);   // A[M=n][k0+2hi, k0+2hi+1]
    v2f b = *(const v2f*)(irow + k0 + 2 * hi);   // B[k0+2hi, k0+2hi+1][N=n]
    c = __builtin_amdgcn_wmma_f32_16x16x4_f32(
        /*neg_a=*/false, a, /*neg_b=*/false, b,
        /*c_mod=*/(short)0, c, /*reuse_a=*/false, /*reuse_b=*/false);
  }

  // final = acc/4 on both sides -> scale dot by 1/16
  const float s = 1.0f / 16.0f;
  int base = wave * 16;
  // C/D layout: VGPR v -> lanes 0-15: (M=v, N=lane); lanes 16-31: (M=v+8, N=lane-16)
  // diag m<8  : VGPR m,   lane m
  // diag m>=8 : VGPR m-8, lane m+16  (i.e. lane = v + 24)
#pragma unroll
  for (int v = 0; v < 8; ++v) {
    if (lane == v      && base + v     < Bn) out[base + v]     = c[v] * s;
    if (lane == v + 24 && base + v + 8 < Bn) out[base + v + 8] = c[v] * s;
  }
}

extern "C" void kernel_launch(void* const* d_in, const int* in_sizes, int n_in,
                              void* d_out, int out_size, void* d_ws, size_t ws_size,
                              hipStream_t stream) {
  const int*   users = (const int*)d_in[0];
  const int*   items = (const int*)d_in[1];
  const int*   edge  = (const int*)d_in[2];
  const float* uw    = (const float*)d_in[3];
  const float* iw    = (const float*)d_in[4];
  float*       out   = (float*)d_out;

  const int Bn = in_sizes[0];
  const int E  = in_sizes[2] / 2;
  const int NU = in_sizes[3] / DIM;
  const int NI = in_sizes[4] / DIM;
  const int N  = NU + NI;
  const size_t total = (size_t)N * DIM;

  const int* row = edge;       // edge_index[0] = src
  const int* col = edge + E;   // edge_index[1] = dst

  // Workspace carve-out (256B aligned regions): norm | deg | cur | nxt | acc
  char* p = (char*)d_ws;
  auto carve = [&](size_t bytes) -> char* {
    char* r = p;
    p += (bytes + 255) & ~(size_t)255;
    return r;
  };
  float* norm = (float*)carve((size_t)E * sizeof(float));
  float* deg  = (float*)carve((size_t)N * sizeof(float));
  float* cur  = (float*)carve(total * sizeof(float));
  float* nxt  = (float*)carve(total * sizeof(float));
  float* acc  = (float*)carve(total * sizeof(float));

  const int T = 256;
  const int total4 = (int)(total / 4);
  const int nu4    = NU * (DIM / 4);
  const int ndeg4  = (N + 3) / 4;

  k_init<<<(total4 + T - 1) / T, T, 0, stream>>>(
      (const float4*)uw, (const float4*)iw,
      (float4*)cur, (float4*)acc, (float4*)nxt, (float4*)deg,
      nu4, total4, ndeg4);
  k_deg<<<(E + T - 1) / T, T, 0, stream>>>(row, deg, E);
  k_dis<<<(N + T - 1) / T, T, 0, stream>>>(deg, N);
  k_norm<<<(E + T - 1) / T, T, 0, stream>>>(row, col, deg, norm, E);

  const int propBlocks = 8192;   // 65536 persistent waves, ~49 edges each
  for (int l = 0; l < 3; ++l) {
    k_prop<<<propBlocks, T, 0, stream>>>(row, col, norm, cur, nxt, E);
    k_fuse<<<(total4 + T - 1) / T, T, 0, stream>>>(
        (float4*)acc, (const float4*)nxt, (float4*)cur, total4);
    float* t = cur; cur = nxt; nxt = t;   // swapped-in nxt is already zeroed
  }

  const int tiles = (Bn + 15) / 16;
  const int dotBlocks = (tiles + 7) / 8;  // 8 waves per 256-thread block
  k_dot<<<dotBlocks, T, 0, stream>>>(acc, users, items, out, NU, Bn);
}